// SimLoss_32272384262254
// MI455X (gfx1250) — compile-verified
//
#include <hip/hip_runtime.h>
#include <cfloat>
#include <math.h>

typedef float v2f __attribute__((ext_vector_type(2)));
typedef float v8f __attribute__((ext_vector_type(8)));

#define NPTS 256
#define DDIM 65536
#define KSPLIT 32
#define KCHUNK (DDIM / KSPLIT) /* 2048 */

// ---------------------------------------------------------------------------
// Zero the GEMM accumulator (d_ws is poisoned once, never re-poisoned).
// ---------------------------------------------------------------------------
__global__ void zero_g(float* __restrict__ G) {
    int i = blockIdx.x * blockDim.x + threadIdx.x;
    if (i < NPTS * NPTS) G[i] = 0.0f;
}

// ---------------------------------------------------------------------------
// Row squared-norms for XA (rows 0..255) and XB (rows 256..511).
// One 256-thread block per row; data is L2-resident after first touch.
// ---------------------------------------------------------------------------
__global__ void row_norms(const float* __restrict__ XA,
                          const float* __restrict__ XB,
                          float* __restrict__ normA,
                          float* __restrict__ normB) {
    __shared__ float red[256];
    const int row = blockIdx.x;
    const float* src = (row < NPTS) ? (XA + (size_t)row * DDIM)
                                    : (XB + (size_t)(row - NPTS) * DDIM);
    const float4* p = (const float4*)src;
    float s = 0.0f;
    for (int i = threadIdx.x; i < DDIM / 4; i += 256) {
        float4 v = p[i];
        s += v.x * v.x + v.y * v.y + v.z * v.z + v.w * v.w;
    }
    red[threadIdx.x] = s;
    __syncthreads();
    for (int off = 128; off > 0; off >>= 1) {
        if (threadIdx.x < (unsigned)off) red[threadIdx.x] += red[threadIdx.x + off];
        __syncthreads();
    }
    if (threadIdx.x == 0) {
        if (row < NPTS) normA[row] = red[0];
        else            normB[row - NPTS] = red[0];
    }
}

// ---------------------------------------------------------------------------
// G += XA * XB^T over one K-chunk. One wave per block (EXEC all-1s for WMMA).
// Each wave owns a 64x64 output tile = 4x4 grid of 16x16 WMMA tiles.
// A-frag / B-frag layout (V_WMMA_F32_16X16X4_F32, 16x4 32-bit operand):
//   lane L holds row (L&15), K pair ((L>>4)*2 .. +1)  -> aligned float2 load.
// ---------------------------------------------------------------------------
__global__ void __launch_bounds__(32)
gemm_wmma(const float* __restrict__ A, const float* __restrict__ B,
          float* __restrict__ G) {
    const int lane = threadIdx.x;          // 0..31, wave32
    const int m0 = blockIdx.x * 64;
    const int n0 = blockIdx.y * 64;
    const int k0 = blockIdx.z * KCHUNK;
    const int r  = lane & 15;
    const int kl = (lane >> 4) * 2;

    v8f acc[4][4];
#pragma unroll
    for (int i = 0; i < 4; ++i)
#pragma unroll
        for (int j = 0; j < 4; ++j)
            acc[i][j] = (v8f)0.0f;

    const float* aB[4];
    const float* bB[4];
#pragma unroll
    for (int t = 0; t < 4; ++t) {
        aB[t] = A + (size_t)(m0 + 16 * t + r) * DDIM + k0 + kl;
        bB[t] = B + (size_t)(n0 + 16 * t + r) * DDIM + k0 + kl;
    }

    for (int kk = 0; kk < KCHUNK; kk += 4) {
        v2f af[4], bf[4];
#pragma unroll
        for (int t = 0; t < 4; ++t) {
            af[t] = *(const v2f*)(aB[t] + kk);
            bf[t] = *(const v2f*)(bB[t] + kk);
        }
#pragma unroll
        for (int i = 0; i < 4; ++i)
#pragma unroll
            for (int j = 0; j < 4; ++j)
                acc[i][j] = __builtin_amdgcn_wmma_f32_16x16x4_f32(
                    /*neg_a=*/false, af[i], /*neg_b=*/false, bf[j],
                    /*c_mod=*/(short)0, acc[i][j],
                    /*reuse_a=*/false, /*reuse_b=*/false);
    }

    // C/D layout: vgpr rr, lane L -> m = rr + (L>>4)*8, n = L&15.
#pragma unroll
    for (int i = 0; i < 4; ++i) {
#pragma unroll
        for (int j = 0; j < 4; ++j) {
            const int n     = n0 + 16 * j + (lane & 15);
            const int mbase = m0 + 16 * i + (lane >> 4) * 8;
#pragma unroll
            for (int rr = 0; rr < 8; ++rr)
                atomicAdd(&G[(size_t)(mbase + rr) * NPTS + n], acc[i][j][rr]);
        }
    }
}

// ---------------------------------------------------------------------------
// Epilogue: d = sqrt(max(||a||^2 + ||b||^2 - 2g, 0)); v = relu(d - tol);
// r1 = mean(min over cols), r2 = mean(min over rows); out = r1 + r2.
// One 256-thread block; thread t handles row t (for r1) and column t (for r2).
// ---------------------------------------------------------------------------
__global__ void finalize(const float* __restrict__ G,
                         const float* __restrict__ nA,
                         const float* __restrict__ nB,
                         const int* __restrict__ tolp,
                         float* __restrict__ out) {
    __shared__ float sr[256];
    __shared__ float sc[256];
    const int t = threadIdx.x;
    const float tol = (float)tolp[0];
    const float na_t = nA[t];
    const float nb_t = nB[t];
    float rmin = FLT_MAX, cmin = FLT_MAX;
    for (int j = 0; j < NPTS; ++j) {
        float sq1 = fmaxf(na_t + nB[j] - 2.0f * G[t * NPTS + j], 0.0f);
        rmin = fminf(rmin, fmaxf(sqrtf(sq1) - tol, 0.0f));
        float sq2 = fmaxf(nA[j] + nb_t - 2.0f * G[j * NPTS + t], 0.0f);
        cmin = fminf(cmin, fmaxf(sqrtf(sq2) - tol, 0.0f));
    }
    sr[t] = rmin;
    sc[t] = cmin;
    __syncthreads();
    for (int off = 128; off > 0; off >>= 1) {
        if (t < off) { sr[t] += sr[t + off]; sc[t] += sc[t + off]; }
        __syncthreads();
    }
    if (t == 0) out[0] = sr[0] / (float)NPTS + sc[0] / (float)NPTS;
}

// ---------------------------------------------------------------------------
extern "C" void kernel_launch(void* const* d_in, const int* in_sizes, int n_in,
                              void* d_out, int out_size, void* d_ws, size_t ws_size,
                              hipStream_t stream) {
    const float* XA  = (const float*)d_in[0];   // [256][65536] f32
    const float* XB  = (const float*)d_in[1];   // [256][65536] f32
    const int*   tol = (const int*)d_in[2];     // scalar int
    float* out = (float*)d_out;                 // 1 float

    float* G  = (float*)d_ws;                   // 256*256
    float* nA = G + NPTS * NPTS;                // 256
    float* nB = nA + NPTS;                      // 256

    zero_g<<<(NPTS * NPTS + 255) / 256, 256, 0, stream>>>(G);
    row_norms<<<2 * NPTS, 256, 0, stream>>>(XA, XB, nA, nB);
    gemm_wmma<<<dim3(4, 4, KSPLIT), 32, 0, stream>>>(XA, XB, G);
    finalize<<<1, 256, 0, stream>>>(G, nA, nB, tol, out);
}